// SpectralLoss_79869211836429
// MI455X (gfx1250) — compile-verified
//
#include <hip/hip_runtime.h>
#include <math.h>

// ---------------------------------------------------------------------------
// SpectralLoss on gfx1250: DFT-as-WMMA-GEMM with async-to-LDS B-panel staging.
//   in : rain_hr float32 (16, 8, 1, 512, 512)
//   out: scalar float loss
// ---------------------------------------------------------------------------

typedef __attribute__((ext_vector_type(2))) float v2f;
typedef __attribute__((ext_vector_type(8))) float v8f;

#define NDIM  512
#define NIMG  16
#define NT    8
#define HW    (NDIM * NDIM)          // 262144
#define ROWS  (NIMG * NDIM)          // 8192 (b,y) rows for the row pass
#define KC    32                     // K-chunk staged in LDS
#define BSTR  40                     // LDS panel row stride (floats): 160B, 16B-aligned,
                                     // +2 rows => +16 banks (mod 64) -> conflict-free frags

// workspace layout (float offsets)
#define WS_MEAN   0                         // 16*512*512
#define WS_FR     (WS_MEAN + NIMG * HW)     // 8192*512
#define WS_FI     (WS_FR   + NIMG * HW)
#define WS_TWC    (WS_FI   + NIMG * HW)     // 512*512
#define WS_TWS    (WS_TWC  + HW)
#define WS_POW    (WS_TWS  + HW)            // 512*512
#define WS_SEG    (WS_POW  + HW)            // 256
#define WS_CNT    (WS_SEG  + 256)           // 256
#define WS_ZERO_N (HW + 512)                // pow + seg + cnt

// ---- WMMA helper (layouts per CDNA5 ISA 7.12.2) ----------------------------
__device__ __forceinline__ v8f wmma4(v2f a, v2f b, v8f c) {
  // D = A(16x4, f32) * B(4x16, f32) + C(16x16, f32)
  return __builtin_amdgcn_wmma_f32_16x16x4_f32(false, a, false, b,
                                               (short)0, c, false, false);
}

// A fragment: lane l -> row m0+(l&15), cols k0+2*(l>>4), k0+2*(l>>4)+1
__device__ __forceinline__ v2f load_a(const float* __restrict__ A,
                                      int m0, int k0, int lo, int hi) {
  const float* p = A + (size_t)(m0 + lo) * NDIM + k0 + (hi << 1);
  v2f r; r.x = p[0]; r.y = p[1]; return r;
}

// B fragment out of an LDS panel: krow already includes the half-wave offset.
__device__ __forceinline__ v2f lds_b(const float* sB, int krow, int col) {
  const float* p = sB + krow * BSTR + col;
  v2f r; r.x = p[0]; r.y = p[BSTR]; return r;
}

// ---- CDNA5 async copy: global -> LDS, 16B per lane, ASYNCcnt tracked -------
__device__ __forceinline__ void async_b128(unsigned lds_off, const float* g) {
  asm volatile("global_load_async_to_lds_b128 %0, %1, off"
               :: "v"(lds_off), "v"(g) : "memory");
}
__device__ __forceinline__ void wait_async0() {
  asm volatile("s_wait_asynccnt 0" ::: "memory");
}
__device__ __forceinline__ unsigned lds_off32(const void* p) {
  // LDS aperture rule: the LDS offset is the address's low 32 bits
  return (unsigned)(uintptr_t)p;
}

// ---- 0) zero the accumulation buffers -------------------------------------
__global__ __launch_bounds__(256) void zero_k(float* __restrict__ ws) {
  int i = blockIdx.x * 256 + threadIdx.x;
  if (i < WS_ZERO_N) ws[WS_POW + i] = 0.0f;
}

// ---- 1) mean over T (streams the whole 134 MB input once) ------------------
__global__ __launch_bounds__(256) void mean_k(const float* __restrict__ in,
                                              float* __restrict__ meanImg) {
  int i = blockIdx.x * 256 + threadIdx.x;          // 0 .. 16*65536-1
  int img = i >> 16;
  int off = i & 65535;
  const float4* base = (const float4*)in + (size_t)img * NT * 65536 + off;
  __builtin_prefetch((const void*)(base + 4 * 65536), 0, 0);  // global_prefetch_b8
  float sx = 0.f, sy = 0.f, sz = 0.f, sw = 0.f;
  #pragma unroll
  for (int t = 0; t < NT; ++t) {
    float4 v = base[(size_t)t * 65536];
    sx += v.x; sy += v.y; sz += v.z; sw += v.w;
  }
  float4 o; o.x = sx * 0.125f; o.y = sy * 0.125f; o.z = sz * 0.125f; o.w = sw * 0.125f;
  ((float4*)meanImg)[i] = o;
}

// ---- 2) twiddle matrices: C[k][n]=cos(2*pi*k*n/512), S=sin (symmetric) -----
__global__ __launch_bounds__(256) void twiddle_k(float* __restrict__ twC,
                                                 float* __restrict__ twS) {
  int i = blockIdx.x * 256 + threadIdx.x;
  if (i >= HW) return;
  int k = i >> 9, n = i & 511;
  int m = (k * n) & 511;                       // periodic in 512
  float ang = (float)m * 1.2271846e-2f;        // 2*pi/512
  twC[i] = cosf(ang);
  twS[i] = sinf(ang);
}

// ---- 3) row pass: Fr = R*C ; Fi = -(R*S) ----------------------------------
// grid (16, 128), block 128 (4 waves). wave tile = 16M x 32N, K=512.
// B-panels (twC/twS, 32x32) staged once per block via async-to-LDS.
__global__ __launch_bounds__(128) void rowdft_k(const float* __restrict__ R,
                                                const float* __restrict__ twC,
                                                const float* __restrict__ twS,
                                                float* __restrict__ Fr,
                                                float* __restrict__ Fi) {
  __shared__ float sC[KC * BSTR];
  __shared__ float sS[KC * BSTR];
  int tid  = threadIdx.x;
  int lane = tid & 31, w = tid >> 5;
  int lo = lane & 15, hi = lane >> 4;
  int m0 = blockIdx.y * 64 + w * 16;           // global (b,y) row
  int n0 = blockIdx.x * 32;                    // output k column
  int rr = tid >> 3;                           // 0..15 : panel row
  int cc = (tid & 7) << 2;                     // float offset within row (16B chunks)
  unsigned lC = lds_off32(sC), lS = lds_off32(sS);
  v8f ar0 = {}, ar1 = {}, ai0 = {}, ai1 = {};
  for (int k0 = 0; k0 < NDIM; k0 += KC) {
    const float* gc = twC + (size_t)(k0 + rr) * NDIM + n0 + cc;
    const float* gs = twS + (size_t)(k0 + rr) * NDIM + n0 + cc;
    async_b128(lC + (unsigned)(rr * BSTR + cc) * 4u,        gc);
    async_b128(lC + (unsigned)((rr + 16) * BSTR + cc) * 4u, gc + 16 * NDIM);
    async_b128(lS + (unsigned)(rr * BSTR + cc) * 4u,        gs);
    async_b128(lS + (unsigned)((rr + 16) * BSTR + cc) * 4u, gs + 16 * NDIM);
    wait_async0();
    __syncthreads();
    #pragma unroll
    for (int kk = 0; kk < KC; kk += 4) {
      v2f a = load_a(R, m0, k0 + kk, lo, hi);
      int krow = kk + (hi << 1);
      v2f bc0 = lds_b(sC, krow, lo);
      v2f bc1 = lds_b(sC, krow, lo + 16);
      v2f bs0 = lds_b(sS, krow, lo);
      v2f bs1 = lds_b(sS, krow, lo + 16);
      ar0 = wmma4(a, bc0, ar0);
      ar1 = wmma4(a, bc1, ar1);
      ai0 = wmma4(a, bs0, ai0);
      ai1 = wmma4(a, bs1, ai1);
    }
    __syncthreads();
  }
  #pragma unroll
  for (int r = 0; r < 8; ++r) {
    size_t m = (size_t)(m0 + r + (hi << 3)) * NDIM;
    Fr[m + n0 + lo]      =  ar0[r];
    Fr[m + n0 + 16 + lo] =  ar1[r];
    Fi[m + n0 + lo]      = -ai0[r];
    Fi[m + n0 + 16 + lo] = -ai1[r];
  }
}

// ---- 4) col pass: Gr = C*Fr + S*Fi ; Gi = C*Fi - S*Fr ; pow += Gr^2+Gi^2 ---
// grid (16, 8, 16), block 128. wave tile = 16M x 32N, image b = blockIdx.z.
// B-panels (Fr/Fi, 32x32) staged once per block via async-to-LDS.
__global__ __launch_bounds__(128) void coldft_k(const float* __restrict__ Fr,
                                                const float* __restrict__ Fi,
                                                const float* __restrict__ twC,
                                                const float* __restrict__ twS,
                                                float* __restrict__ powhw) {
  __shared__ float sFr[KC * BSTR];
  __shared__ float sFi[KC * BSTR];
  int tid  = threadIdx.x;
  int lane = tid & 31, w = tid >> 5;
  int lo = lane & 15, hi = lane >> 4;
  int b  = blockIdx.z;
  const float* FrB = Fr + (size_t)b * HW;
  const float* FiB = Fi + (size_t)b * HW;
  int m0 = blockIdx.y * 64 + w * 16;           // output frequency row m
  int n0 = blockIdx.x * 32;                    // output frequency col k
  int rr = tid >> 3;
  int cc = (tid & 7) << 2;
  unsigned lR = lds_off32(sFr), lI = lds_off32(sFi);
  v8f cr0 = {}, cr1 = {}, si0 = {}, si1 = {};  // C*Fr , S*Fi
  v8f ci0 = {}, ci1 = {}, sr0 = {}, sr1 = {};  // C*Fi , S*Fr
  for (int k0 = 0; k0 < NDIM; k0 += KC) {      // K dim = y
    const float* gr = FrB + (size_t)(k0 + rr) * NDIM + n0 + cc;
    const float* gi = FiB + (size_t)(k0 + rr) * NDIM + n0 + cc;
    async_b128(lR + (unsigned)(rr * BSTR + cc) * 4u,        gr);
    async_b128(lR + (unsigned)((rr + 16) * BSTR + cc) * 4u, gr + 16 * NDIM);
    async_b128(lI + (unsigned)(rr * BSTR + cc) * 4u,        gi);
    async_b128(lI + (unsigned)((rr + 16) * BSTR + cc) * 4u, gi + 16 * NDIM);
    wait_async0();
    __syncthreads();
    #pragma unroll
    for (int kk = 0; kk < KC; kk += 4) {
      v2f ac = load_a(twC, m0, k0 + kk, lo, hi);
      v2f as = load_a(twS, m0, k0 + kk, lo, hi);
      int krow = kk + (hi << 1);
      v2f br0 = lds_b(sFr, krow, lo);
      v2f br1 = lds_b(sFr, krow, lo + 16);
      v2f bi0 = lds_b(sFi, krow, lo);
      v2f bi1 = lds_b(sFi, krow, lo + 16);
      cr0 = wmma4(ac, br0, cr0);  cr1 = wmma4(ac, br1, cr1);
      si0 = wmma4(as, bi0, si0);  si1 = wmma4(as, bi1, si1);
      ci0 = wmma4(ac, bi0, ci0);  ci1 = wmma4(ac, bi1, ci1);
      sr0 = wmma4(as, br0, sr0);  sr1 = wmma4(as, br1, sr1);
    }
    __syncthreads();
  }
  #pragma unroll
  for (int r = 0; r < 8; ++r) {
    size_t m = (size_t)(m0 + r + (hi << 3)) * NDIM;
    float gr = cr0[r] + si0[r];
    float gi = ci0[r] - sr0[r];
    atomicAdd(&powhw[m + n0 + lo], gr * gr + gi * gi);
    gr = cr1[r] + si1[r];
    gi = ci1[r] - sr1[r];
    atomicAdd(&powhw[m + n0 + 16 + lo], gr * gr + gi * gi);
  }
}

// ---- 5) radial binning with LDS bins --------------------------------------
__global__ __launch_bounds__(256) void radial_k(const float* __restrict__ powhw,
                                                float* __restrict__ seg,
                                                float* __restrict__ cnt) {
  __shared__ float sSeg[256];
  __shared__ float sCnt[256];
  int tid = threadIdx.x;
  sSeg[tid] = 0.0f; sCnt[tid] = 0.0f;
  __syncthreads();
  int i = blockIdx.x * 256 + tid;              // 1024 blocks cover 512x512
  int y = i >> 9, x = i & 511;
  int dx = x - 256, dy = y - 256;
  int r2 = dx * dx + dy * dy;
  int r = (int)sqrtf((float)r2);
  while ((r + 1) * (r + 1) <= r2) ++r;         // exact floor(sqrt(r2))
  while (r * r > r2) --r;
  if (r < 256) {
    atomicAdd(&sSeg[r], powhw[i]);
    atomicAdd(&sCnt[r], 1.0f);
  }
  __syncthreads();
  atomicAdd(&seg[tid], sSeg[tid]);
  atomicAdd(&cnt[tid], sCnt[tid]);
}

// ---- 6) final normalization + MSE (single block) ---------------------------
__global__ __launch_bounds__(256) void loss_k(const float* __restrict__ seg,
                                              const float* __restrict__ cnt,
                                              float* __restrict__ out) {
  __shared__ float red[256];
  int tid = threadIdx.x;
  float gen = 0.0f, refv = 0.0f;
  if (tid >= 1) {                              // bins 1..255
    float c = cnt[tid];
    float denom = fmaxf(c, 1.0f) * (float)NIMG;  // B*C = 16
    gen  = (c > 0.0f) ? seg[tid] / denom : 0.0f;
    refv = powf((float)tid, -5.0f / 3.0f);
  }
  red[tid] = gen; __syncthreads();
  for (int s = 128; s > 0; s >>= 1) { if (tid < s) red[tid] += red[tid + s]; __syncthreads(); }
  float sum_gen = red[0]; __syncthreads();
  red[tid] = refv; __syncthreads();
  for (int s = 128; s > 0; s >>= 1) { if (tid < s) red[tid] += red[tid + s]; __syncthreads(); }
  float sum_ref = red[0]; __syncthreads();
  float d = 0.0f;
  if (tid >= 1) {
    float gn = gen  / (sum_gen + 1e-8f);
    float rn = refv / (sum_ref + 1e-8f);
    float e = gn - rn;
    d = e * e;
  }
  red[tid] = d; __syncthreads();
  for (int s = 128; s > 0; s >>= 1) { if (tid < s) red[tid] += red[tid + s]; __syncthreads(); }
  if (tid == 0) out[0] = red[0] / 255.0f;
}

// ---------------------------------------------------------------------------
extern "C" void kernel_launch(void* const* d_in, const int* in_sizes, int n_in,
                              void* d_out, int out_size, void* d_ws, size_t ws_size,
                              hipStream_t stream) {
  (void)in_sizes; (void)n_in; (void)out_size; (void)ws_size;
  const float* in = (const float*)d_in[0];
  float* out = (float*)d_out;
  float* ws  = (float*)d_ws;

  float* meanImg = ws + WS_MEAN;
  float* Fr      = ws + WS_FR;
  float* Fi      = ws + WS_FI;
  float* twC     = ws + WS_TWC;
  float* twS     = ws + WS_TWS;
  float* powhw   = ws + WS_POW;
  float* seg     = ws + WS_SEG;
  float* cnt     = ws + WS_CNT;

  zero_k<<<(WS_ZERO_N + 255) / 256, 256, 0, stream>>>(ws);
  mean_k<<<(NIMG * HW / 4 + 255) / 256, 256, 0, stream>>>(in, meanImg);
  twiddle_k<<<(HW + 255) / 256, 256, 0, stream>>>(twC, twS);

  {
    dim3 grid(NDIM / 32, ROWS / 64, 1);        // (16, 128)
    rowdft_k<<<grid, 128, 0, stream>>>(meanImg, twC, twS, Fr, Fi);
  }
  {
    dim3 grid(NDIM / 32, NDIM / 64, NIMG);     // (16, 8, 16)
    coldft_k<<<grid, 128, 0, stream>>>(Fr, Fi, twC, twS, powhw);
  }
  radial_k<<<HW / 256, 256, 0, stream>>>(powhw, seg, cnt);
  loss_k<<<1, 256, 0, stream>>>(seg, cnt, out);
}